// ScaledDotProductAttention_65867618452297
// MI455X (gfx1250) — compile-verified
//
#include <hip/hip_runtime.h>
#include <hip/hip_bf16.h>

typedef __attribute__((ext_vector_type(2))) float v2f;
typedef __attribute__((ext_vector_type(4))) float v4f;
typedef __attribute__((ext_vector_type(8))) float v8f;

static constexpr int B_SZ = 1024;
static constexpr int T_SZ = 1024;
static constexpr int D_SZ = 512;
static constexpr int NWAVE = 8;                       // 256 threads / wave32
static constexpr int NTILE = (T_SZ / 16) / NWAVE;     // tiles per wave
static constexpr float QSCALE = 0.04419417382415922f; // 1/sqrt(512)

__global__ __launch_bounds__(256) void sdpa_flash_wmma_kernel(
    const float* __restrict__ q,     // [B, D]
    const float* __restrict__ v,     // [B, T, D]
    float* __restrict__ ctx_out,     // [B, D]
    float* __restrict__ attn_out)    // [B, T]
{
    __shared__ float qs[D_SZ];                 // scaled query row
    __shared__ float sc[T_SZ];                 // raw scores for this b
    __shared__ float red_m[NWAVE];
    __shared__ float red_l[NWAVE];
    __shared__ float red_acc[NWAVE][D_SZ];

    const int b    = blockIdx.x;
    const int tid  = threadIdx.x;
    const int wave = tid >> 5;
    const int lane = tid & 31;
    const int row  = lane & 15;   // A-matrix row (t within tile)
    const int hi   = lane >> 4;   // K half-select for 16x4 A/B layouts

    // ---- stage q/sqrt(D) into LDS ----
    {
        const float2* qg = (const float2*)(q + (size_t)b * D_SZ);
        float2 qv = qg[tid];                    // 256 threads * 2 floats = 512
        qs[2 * tid]     = qv.x * QSCALE;
        qs[2 * tid + 1] = qv.y * QSCALE;
    }
    __syncthreads();

    const float* __restrict__ vb = v + (size_t)b * T_SZ * D_SZ;

    // Prefetch this wave's first tile (256 x 128B lines over 8 wave-ops).
    {
        const float* pf = vb + (size_t)(wave * 16 + row) * D_SZ + hi * 32;
        #pragma unroll
        for (int j = 0; j < 8; ++j)
            __builtin_prefetch(pf + j * 64, 0, 3);
    }

    float m = -INFINITY;
    float l = 0.0f;
    v4f acc0 = {0.f, 0.f, 0.f, 0.f};           // d = 4*lane + 0..3
    v4f acc1 = {0.f, 0.f, 0.f, 0.f};           // d = 4*lane + 128..131
    v4f acc2 = {0.f, 0.f, 0.f, 0.f};           // d = 4*lane + 256..259
    v4f acc3 = {0.f, 0.f, 0.f, 0.f};           // d = 4*lane + 384..387

    for (int it = 0; it < NTILE; ++it) {
        const int t0 = (wave + it * NWAVE) * 16;

        // ---------- phase 1: scores for 16 t rows via f32 WMMA ----------
        // A (16x4): lane(0-15)=row m, VGPR0/1 = K0/K1; lanes 16-31 = K2/K3.
        // B (4x16): q chunk broadcast to every column (mirrored lane layout).
        v8f c = {0.f, 0.f, 0.f, 0.f, 0.f, 0.f, 0.f, 0.f};
        const float* arow = vb + (size_t)(t0 + row) * D_SZ + 2 * hi;
        const float* qrow = qs + 2 * hi;
        #pragma unroll 8
        for (int k0 = 0; k0 < D_SZ; k0 += 4) {
            v2f a  = *(const v2f*)(arow + k0);   // global_load_b64 (WGP$-hot)
            v2f bq = *(const v2f*)(qrow + k0);   // ds_load_b64
            c = __builtin_amdgcn_wmma_f32_16x16x4_f32(
                    false, a, false, bq, (short)0, c, false, false);
        }

        // All columns of D are identical: lane 0 holds rows 0-7 (c[0..7]),
        // lane 16 holds rows 8-15. Park raw scores in LDS.
        if (row == 0) {
            #pragma unroll
            for (int r = 0; r < 8; ++r) sc[t0 + 8 * hi + r] = c[r];
        }

        // Prefetch the NEXT tile while phase 2 chews on this one.
        if (it + 1 < NTILE) {
            const int tn = (wave + (it + 1) * NWAVE) * 16;
            const float* pf = vb + (size_t)(tn + row) * D_SZ + hi * 32;
            #pragma unroll
            for (int j = 0; j < 8; ++j)
                __builtin_prefetch(pf + j * 64, 0, 3);
        }

        // ---------- online softmax update ----------
        float pm = c[0];
        #pragma unroll
        for (int r = 1; r < 8; ++r) pm = fmaxf(pm, c[r]);
        float mt    = fmaxf(__shfl(pm, 0, 32), __shfl(pm, 16, 32));
        float m_new = fmaxf(m, mt);
        float resc  = __expf(m - m_new);   // 0 on first tile (m = -inf)
        l *= resc;
        acc0 *= resc; acc1 *= resc; acc2 *= resc; acc3 *= resc;

        const float* vrow = vb + (size_t)t0 * D_SZ + 4 * lane;
        #pragma unroll
        for (int t = 0; t < 16; ++t) {
            float s_t = __shfl(c[t & 7], (t >> 3) << 4, 32);
            float p   = __expf(s_t - m_new);
            l += p;
            const v4f* vp = (const v4f*)(vrow + (size_t)t * D_SZ);
            acc0 += p * vp[0];      // fully coalesced b128 loads
            acc1 += p * vp[32];
            acc2 += p * vp[64];
            acc3 += p * vp[96];
        }
        m = m_new;
    }

    // ---------- combine the 8 waves ----------
    if (lane == 0) { red_m[wave] = m; red_l[wave] = l; }
    {
        float* ra = &red_acc[wave][4 * lane];
        *(v4f*)(ra)       = acc0;
        *(v4f*)(ra + 128) = acc1;
        *(v4f*)(ra + 256) = acc2;
        *(v4f*)(ra + 384) = acc3;
    }
    __syncthreads();

    float M = red_m[0];
    #pragma unroll
    for (int w = 1; w < NWAVE; ++w) M = fmaxf(M, red_m[w]);
    float e[NWAVE];
    float Z = 0.0f;
    #pragma unroll
    for (int w = 0; w < NWAVE; ++w) {
        e[w] = __expf(red_m[w] - M);
        Z += red_l[w] * e[w];
    }
    const float invZ = 1.0f / Z;

    // context vector: each thread owns d = tid and d = tid + 256
    #pragma unroll
    for (int d = tid; d < D_SZ; d += 256) {
        float s = 0.0f;
        #pragma unroll
        for (int w = 0; w < NWAVE; ++w) s += red_acc[w][d] * e[w];
        ctx_out[(size_t)b * D_SZ + d] = s * invZ;
    }

    // attention weights: normalize raw scores from LDS, coalesced stores
    float* ap = attn_out + (size_t)b * T_SZ;
    #pragma unroll
    for (int t = tid; t < T_SZ; t += 256)
        ap[t] = __expf(sc[t] - M) * invZ;
}

extern "C" void kernel_launch(void* const* d_in, const int* in_sizes, int n_in,
                              void* d_out, int out_size, void* d_ws, size_t ws_size,
                              hipStream_t stream) {
    const float* q = (const float*)d_in[0];   // [B, D]
    const float* v = (const float*)d_in[1];   // [B, T, D]
    float* ctx  = (float*)d_out;                          // [B, D]
    float* attn = (float*)d_out + (size_t)B_SZ * D_SZ;    // [B, T, 1]
    hipLaunchKernelGGL(sdpa_flash_wmma_kernel, dim3(B_SZ), dim3(256), 0, stream,
                       q, v, ctx, attn);
}